// TensorRelationalGraphConvolution_77386720740028
// MI455X (gfx1250) — compile-verified
//
#include <hip/hip_runtime.h>

typedef __attribute__((ext_vector_type(2))) float v2f;
typedef __attribute__((ext_vector_type(8))) float v8f;

// Problem constants (from the reference)
#define D_DIM   128      // D_IN == D_OUT
#define NREL    17       // NUM_REL + 1
#define NSAMP   32
#define ROWS    16       // batch rows per workgroup (one M-tile)
#define KDIM    (NREL * D_DIM)   // 2176 flattened K
#define KSTRIDE (KDIM + 4)       // pad 4 dwords -> stride mod 64 == 4, bank-conflict free

__global__ __launch_bounds__(512, 1)
void rgcn_fused_wmma_kernel(const float* __restrict__ embed,   // [1M, 128]
                            const float* __restrict__ W,       // [17, 128, 128] (r, o, i)
                            const int*   __restrict__ nbr,     // [B, 32]
                            const int*   __restrict__ rel,     // [B, 32]
                            float* __restrict__ out)           // [B, 128]
{
    extern __shared__ float smem[];   // ROWS * KSTRIDE floats = 139,520 B

    const int tid  = threadIdx.x;
    const int wave = tid >> 5;
    const int lane = tid & 31;

    // ---------------- Phase 1: gather + per-relation aggregate ----------------
    // Wave w owns batch row n; lane owns dims [4*lane, 4*lane+4).
    {
        const int n = blockIdx.x * ROWS + wave;
        float* aggRow = smem + wave * KSTRIDE;

        // zero this row's 17*128 accumulator
        #pragma unroll
        for (int r = 0; r < NREL; ++r) {
            float4 z = make_float4(0.f, 0.f, 0.f, 0.f);
            *(float4*)(aggRow + r * D_DIM + lane * 4) = z;
        }

        const int* nbrRow = nbr + (size_t)n * NSAMP;
        const int* relRow = rel + (size_t)n * NSAMP;

        for (int s = 0; s < NSAMP; ++s) {
            const int nb = nbrRow[s];            // uniform across wave
            const int rr = relRow[s];            // uniform across wave
            const float4 e = *(const float4*)(embed + (size_t)nb * D_DIM + lane * 4);
            float* dst = aggRow + rr * D_DIM + lane * 4;
            float4 a = *(float4*)dst;
            a.x += e.x; a.y += e.y; a.z += e.z; a.w += e.w;
            *(float4*)dst = a;                   // lanes own disjoint dims; wave-serial -> no atomics
        }
    }

    __syncthreads();

    // ---------------- Phase 2: C[16x128] = A[16x2176] x B[2176x128] ----------------
    // Waves 0..7 each own one 16-wide N-tile; full K per wave (544 WMMAs).
    if (wave < 8) {
        const int ntile = wave;
        const int mrow  = lane & 15;     // A: M index / B,C: N index (within tile)
        const int khalf = lane >> 4;     // 0: K=k0,k0+1 ; 1: K=k0+2,k0+3

        // A element [M=mrow, K=k0+2*khalf (+1)] from padded LDS
        const float* aBase = smem + mrow * KSTRIDE + 2 * khalf;
        // B element [K, N] = W[r, n, i] with k=(r,i) flattened; n = ntile*16 + mrow
        const float* bBase = W + (size_t)(ntile * 16 + mrow) * D_DIM + 2 * khalf;

        v8f acc = {};
        #pragma unroll 8
        for (int k0 = 0; k0 < KDIM; k0 += 4) {
            const int r = k0 >> 7;       // k0 / 128
            const int i = k0 & 127;      // k0 % 128
            v2f a = *(const v2f*)(aBase + k0);
            v2f b = *(const v2f*)(bBase + (size_t)r * (D_DIM * D_DIM) + i);
            // D = A x B + C  (16x16x4, f32)
            acc = __builtin_amdgcn_wmma_f32_16x16x4_f32(
                      /*neg_a=*/false, a, /*neg_b=*/false, b,
                      /*c_mod=*/(short)0, acc, /*reuse_a=*/false, /*reuse_b=*/false);
        }

        // fold mean(/32), mean(/17) and *1000 into one scale; ReLU; store
        const float scale = 1000.0f / (32.0f * 17.0f);
        float* outBase = out + (size_t)blockIdx.x * ROWS * D_DIM + ntile * 16 + mrow;
        #pragma unroll
        for (int v = 0; v < 8; ++v) {
            const int M = v + (khalf ? 8 : 0);     // C/D layout: lanes 16-31 hold M+8
            float val = acc[v] * scale;
            val = val > 0.0f ? val : 0.0f;
            outBase[(size_t)M * D_DIM] = val;
        }
    }
}

extern "C" void kernel_launch(void* const* d_in, const int* in_sizes, int n_in,
                              void* d_out, int out_size, void* d_ws, size_t ws_size,
                              hipStream_t stream) {
    const float* embed = (const float*)d_in[0];   // embed_table      [1M * 128] f32
    const float* W     = (const float*)d_in[1];   // relation_weights [17*128*128] f32
    const int*   nbr   = (const int*)d_in[2];     // sampled_neighbors [B*32]
    const int*   rel   = (const int*)d_in[3];     // sampled_relations [B*32]
    float* out = (float*)d_out;                   // [B * 128] f32

    const int batch = in_sizes[2] / NSAMP;        // 16384
    const dim3 grid(batch / ROWS);                // 1024 workgroups
    const dim3 block(512);                        // 16 waves (wave32)
    const size_t ldsBytes = (size_t)ROWS * KSTRIDE * sizeof(float);  // 139,520 B

    hipLaunchKernelGGL(rgcn_fused_wmma_kernel, grid, block, ldsBytes, stream,
                       embed, W, nbr, rel, out);
}